// AdditiveAttention_6459630813189
// MI455X (gfx1250) — compile-verified
//
#include <hip/hip_runtime.h>
#include <hip/hip_bf16.h>

// Problem constants (from reference)
constexpr int BB   = 4;
constexpr int QQ   = 512;
constexpr int KK   = 2048;
constexpr int DD   = 256;
constexpr int HH   = 64;
constexpr float NEGV = -1000000.0f;

typedef __attribute__((ext_vector_type(2))) float v2f;
typedef __attribute__((ext_vector_type(8))) float v8f;
typedef __attribute__((ext_vector_type(4))) int   v4i;

// ---------------------------------------------------------------------------
// Async global->LDS staging (gfx1250 GLOBAL_LOAD_ASYNC_TO_LDS_B128, ASYNCcnt)
// with a synchronous fallback if the builtin is absent on this toolchain.
// Builtin signature (from hipcc diagnostic): param 1 is a non-const
// v4i pointer in the global (__device__/AS1) address space.
// ---------------------------------------------------------------------------
#if __has_builtin(__builtin_amdgcn_global_load_async_to_lds_b128)
#define HAVE_ASYNC_LDS 1
#else
#define HAVE_ASYNC_LDS 0
#endif

__device__ __forceinline__ void lds_copy16(float* dst_lds, const float* src_global) {
#if HAVE_ASYNC_LDS
  __builtin_amdgcn_global_load_async_to_lds_b128(
      (__attribute__((address_space(1))) v4i*)src_global,
      (__attribute__((address_space(3))) v4i*)dst_lds,
      /*imm offset=*/0, /*cpol=*/0);
#else
  *(float4*)dst_lds = *(const float4*)src_global;
#endif
}

__device__ __forceinline__ void lds_copy_wait() {
#if HAVE_ASYNC_LDS
#if __has_builtin(__builtin_amdgcn_s_wait_asynccnt)
  __builtin_amdgcn_s_wait_asynccnt(0);
#else
  asm volatile("s_wait_asynccnt 0" ::: "memory");
#endif
#endif
  __syncthreads();
}

// ---------------------------------------------------------------------------
// fp32 WMMA GEMM, strip form: each wave owns a 16x(16*NT) strip of C and
// holds NT f32 accumulators, so the A fragment is loaded once per k-step and
// feeds NT back-to-back V_WMMA_F32_16X16X4_F32 ops (A L2 traffic / NT).
// Pointer-increment addressing keeps the inner loop free of 64-bit muls.
// Batched via blockIdx.z strides.
// ---------------------------------------------------------------------------
template <int NT>
__global__ void gemm_strip_f32_wmma(const float* __restrict__ A,
                                    const float* __restrict__ Bm,
                                    float* __restrict__ C,
                                    int M, int N, int Kd,
                                    long sA, long sB, long sC) {
  A  += (size_t)blockIdx.z * (size_t)sA;
  Bm += (size_t)blockIdx.z * (size_t)sB;
  C  += (size_t)blockIdx.z * (size_t)sC;

  const int lane   = threadIdx.x & 31;
  const int wave   = (blockIdx.x * blockDim.x + threadIdx.x) >> 5;
  const int strips = N / (16 * NT);
  const int tm     = (wave / strips) << 4;
  const int tn0    = (wave % strips) * 16 * NT;
  if (tm >= M) return;                  // wave-uniform: EXEC stays all-ones

  const int half = lane >> 4;           // which K pair this half-wave owns
  const int l16  = lane & 15;

  v8f acc[NT];
  #pragma unroll
  for (int t = 0; t < NT; ++t) acc[t] = v8f{};

  // A fragment: row tm+l16, K = kk + 2*half + {0,1}
  const float* aptr = A + (size_t)(tm + l16) * Kd + 2 * half;
  // B fragment: row kk + 2*half + {0,1}, col tn0 + t*16 + l16
  const float* bptr = Bm + (size_t)(2 * half) * N + tn0 + l16;

  for (int kk = 0; kk < Kd; kk += 4) {
    v2f a;
    a.x = aptr[0];
    a.y = aptr[1];
    #pragma unroll
    for (int t = 0; t < NT; ++t) {
      v2f b;
      b.x = bptr[t * 16];
      b.y = bptr[t * 16 + N];
      acc[t] = __builtin_amdgcn_wmma_f32_16x16x4_f32(
          false, a, false, b, (short)0, acc[t], false, false);
    }
    aptr += 4;
    bptr += (size_t)4 * N;
  }

  float* crow = C + (size_t)(tm + 8 * half) * N + tn0 + l16;
  #pragma unroll
  for (int t = 0; t < NT; ++t)
    #pragma unroll
    for (int j = 0; j < 8; ++j)
      crow[(size_t)j * N + t * 16] = acc[t][j];
}

// ---------------------------------------------------------------------------
// Fused additive-attention scores (dominant stage: 268M hardware tanh).
//   scores[b,q,k] = sum_h wv[h]*tanh(qh[b,q,h]+kh[b,k,h]), masked.
// Tile: 16 q-rows x 128 k-rows per block (256 threads). kh tile (33 KB,
// padded rows for conflict-free column access) staged in LDS via async
// b128 loads; each thread owns one k and 8 q's -> every LDS read feeds
// 8 tanh+fma, so the loop runs on the trans pipe, not memory.
// kh L2 traffic drops from ~1 GB to ~64 MB.
// ---------------------------------------------------------------------------
constexpr int QT  = 16;   // q rows per block
constexpr int KT  = 128;  // k rows per block
constexpr int QPT = 8;    // q rows per thread (QT*KT/256)

__device__ __forceinline__ float fast_tanh(float x) {
#if __has_builtin(__builtin_amdgcn_tanhf)
  return __builtin_amdgcn_tanhf(x);            // v_tanh_f32 (gfx1250 trans op)
#else
  float e = __builtin_amdgcn_exp2f(x * 2.8853900817779268f);
  return 1.0f - 2.0f * __builtin_amdgcn_rcpf(e + 1.0f);
#endif
}

__global__ void scores_tanh_kernel(const float* __restrict__ qh,
                                   const float* __restrict__ kh,
                                   const float* __restrict__ wv,
                                   const long long* __restrict__ valid_lens,
                                   float* __restrict__ scores) {
  const int b   = blockIdx.z;
  const int q0  = blockIdx.y * QT;
  const int k0  = blockIdx.x * KT;
  const int tid = threadIdx.x;          // 256
  const int kl  = tid & (KT - 1);       // local k (0..127)
  const int qg  = tid >> 7;             // 0|1: which 8-q group

  __shared__ float khs[KT][HH + 1];     // +1 pad: column reads conflict-free
  __shared__ float qs[QT][HH];
  __shared__ float wvs[HH];

  // --- stage kh tile (KT*HH = 8192 floats, contiguous in global) ---
  const float* ksrc = kh + ((size_t)b * KK + k0) * HH;
  #pragma unroll
  for (int i = 0; i < (KT * HH / 4) / 256; ++i) {   // 8 x b128 per thread
    const int f   = i * 256 + tid;                  // float4 index in tile
    const int row = f >> 4;                         // 16 float4 per row
    const int c4  = f & 15;
    lds_copy16(&khs[row][c4 * 4], ksrc + (size_t)f * 4);
  }
  // --- stage q rows (QT*HH = 1024 floats) + wv ---
  const float* qsrc = qh + ((size_t)b * QQ + q0) * HH;
  {
    const int row = tid >> 4, c4 = tid & 15;
    lds_copy16(&qs[row][c4 * 4], qsrc + (size_t)tid * 4);
  }
  if (tid < HH) wvs[tid] = wv[tid];
  lds_copy_wait();                      // s_wait_asynccnt 0 + barrier

  float acc[QPT] = {};
  #pragma unroll
  for (int h = 0; h < HH; ++h) {
    const float kv  = khs[kl][h];
    const float wvh = wvs[h];
    #pragma unroll
    for (int j = 0; j < QPT; ++j)
      acc[j] = fmaf(wvh, fast_tanh(qs[qg * QPT + j][h] + kv), acc[j]);
  }

  const long long vl = valid_lens[b];
  const bool valid   = (long long)(k0 + kl) < vl;
  float* srow = scores + ((size_t)b * QQ + q0 + qg * QPT) * KK + k0 + kl;
  #pragma unroll
  for (int j = 0; j < QPT; ++j)
    srow[(size_t)j * KK] = valid ? acc[j] : NEGV;
}

// ---------------------------------------------------------------------------
// In-place masked softmax over K=2048 per (b,q) row. 8 waves: wave32 shuffle
// reduction + LDS cross-wave combine.
// ---------------------------------------------------------------------------
__global__ void softmax_rows_kernel(float* __restrict__ sc) {
  const int q   = blockIdx.x;
  const int b   = blockIdx.y;
  const int tid = threadIdx.x;
  float* row = sc + ((size_t)b * QQ + q) * KK;

  __shared__ float red[8];

  float m = -3.402823e38f;
  #pragma unroll
  for (int k = tid; k < KK; k += 256) m = fmaxf(m, row[k]);
  #pragma unroll
  for (int off = 16; off > 0; off >>= 1) m = fmaxf(m, __shfl_xor(m, off, 32));
  if ((tid & 31) == 0) red[tid >> 5] = m;
  __syncthreads();
  m = red[0];
  #pragma unroll
  for (int w = 1; w < 8; ++w) m = fmaxf(m, red[w]);
  __syncthreads();

  float sum = 0.0f;
  #pragma unroll
  for (int k = tid; k < KK; k += 256) {
    float e = __builtin_amdgcn_exp2f((row[k] - m) * 1.4426950408889634f);
    row[k] = e;
    sum += e;
  }
  #pragma unroll
  for (int off = 16; off > 0; off >>= 1) sum += __shfl_xor(sum, off, 32);
  if ((tid & 31) == 0) red[tid >> 5] = sum;
  __syncthreads();
  sum = 0.0f;
  #pragma unroll
  for (int w = 0; w < 8; ++w) sum += red[w];

  const float inv = __builtin_amdgcn_rcpf(sum);
  #pragma unroll
  for (int k = tid; k < KK; k += 256) row[k] *= inv;
}

// ---------------------------------------------------------------------------
// Host-side orchestration
// ---------------------------------------------------------------------------
extern "C" void kernel_launch(void* const* d_in, const int* in_sizes, int n_in,
                              void* d_out, int out_size, void* d_ws, size_t ws_size,
                              hipStream_t stream) {
  const float*     queries    = (const float*)d_in[0];     // (B,Q,D)
  const float*     keys       = (const float*)d_in[1];     // (B,K,D)
  const float*     values     = (const float*)d_in[2];     // (B,K,D)
  const long long* valid_lens = (const long long*)d_in[3]; // (B,) int64
  const float*     Wq         = (const float*)d_in[4];     // (D,H)
  const float*     Wk         = (const float*)d_in[5];     // (D,H)
  const float*     wv         = (const float*)d_in[6];     // (H,)
  float*           out        = (float*)d_out;             // (B,Q,D)

  // Workspace: qh | kh | scores  (~19.4 MB fp32)
  float* qh = (float*)d_ws;                 // B*Q*H
  float* kh = qh + (size_t)BB * QQ * HH;    // B*K*H
  float* sc = kh + (size_t)BB * KK * HH;    // B*Q*K

  // 1) qh = (B*Q, D) @ (D, H); N=64 -> one 4-tile strip per wave: 128 waves
  {
    const int M = BB * QQ, N = HH, Kd = DD;
    const int waves = (M / 16) * (N / 64);
    gemm_strip_f32_wmma<4><<<dim3(waves / 8, 1, 1), 256, 0, stream>>>(
        queries, Wq, qh, M, N, Kd, 0, 0, 0);
  }
  // 2) kh = (B*K, D) @ (D, H): 512 waves
  {
    const int M = BB * KK, N = HH, Kd = DD;
    const int waves = (M / 16) * (N / 64);
    gemm_strip_f32_wmma<4><<<dim3(waves / 8, 1, 1), 256, 0, stream>>>(
        keys, Wk, kh, M, N, Kd, 0, 0, 0);
  }
  // 3) scores + tanh + mask (dominant stage)
  scores_tanh_kernel<<<dim3(KK / KT, QQ / QT, BB), 256, 0, stream>>>(
      qh, kh, wv, valid_lens, sc);
  // 4) softmax over K, in place
  softmax_rows_kernel<<<dim3(QQ, BB), 256, 0, stream>>>(sc);
  // 5) out[b] = attn[b] (Q,K) @ values[b] (K,D): 16x64 strips, 128 waves/batch
  {
    const int M = QQ, N = DD, Kd = KK;
    const int waves = (M / 16) * (N / 64);
    gemm_strip_f32_wmma<4><<<dim3(waves / 8, 1, BB), 256, 0, stream>>>(
        sc, values, out, M, N, Kd,
        (long)QQ * KK, (long)KK * DD, (long)QQ * DD);
  }
}